// Point2Encoder_29892972380431
// MI455X (gfx1250) — compile-verified
//
#include <hip/hip_runtime.h>
#include <cstddef>

// ---------------------------------------------------------------------------
// PointNet++ MSG encoder for MI455X (gfx1250, wave32).
// - All GEMMs: V_WMMA_F32_16X16X4_F32 (fp32 precision; workload is ~27 GFLOP
//   and entirely L2-resident -> latency bound, so fp32 WMMA is the right
//   precision/throughput point).
// - Point-cloud staging and neighbor gathers use GLOBAL_LOAD_ASYNC_TO_LDS
//   (ASYNCcnt-tracked direct memory->LDS copies).
// - FPS keeps the whole cloud + running dmin in the 320KB/WGP LDS.
// ---------------------------------------------------------------------------

typedef float v2f __attribute__((ext_vector_type(2)));
typedef float v8f __attribute__((ext_vector_type(8)));

typedef __attribute__((address_space(1))) int gas_int;  // global AS
typedef __attribute__((address_space(3))) int las_int;  // LDS AS

#define KMAX 32

__device__ __forceinline__ v8f wmma_f32_4(v2f a, v2f b, v8f c) {
  // 8 args: (neg_a, A, neg_b, B, c_mod, C, reuse_a, reuse_b)
  return __builtin_amdgcn_wmma_f32_16x16x4_f32(false, a, false, b, (short)0, c,
                                               false, false);
}

// Direct memory->LDS async copy (CDNA5 GLOBAL_LOAD_ASYNC_TO_LDS_B32,
// ASYNCcnt-tracked). Builtin signature (from the compiler diagnostic):
// (as1 int* gsrc, as3 int* ldst, imm int offset, imm int cpol).
__device__ __forceinline__ void async_copy_b32(const float* gsrc, float* ldst) {
#if __has_builtin(__builtin_amdgcn_global_load_async_to_lds_b32)
  __builtin_amdgcn_global_load_async_to_lds_b32((gas_int*)gsrc, (las_int*)ldst,
                                                0, 0);
#else
  *ldst = *gsrc;
#endif
}

__device__ __forceinline__ void wait_async() {
#if __has_builtin(__builtin_amdgcn_s_wait_asynccnt)
  __builtin_amdgcn_s_wait_asynccnt(0);
#endif
}

// ---------------------------------------------------------------------------
// 1) Feature projection: feat0[r][c] = xyz[r][:] @ proj_W + proj_b
//    M = B*N rows, K = 3 (padded to 4), Ncols = 64.
//    One wave per 16x16 output tile; block = 256 thr (8 waves) = 32 rows.
// ---------------------------------------------------------------------------
__global__ void proj_kernel(const float* __restrict__ xyz,
                            const float* __restrict__ W,
                            const float* __restrict__ bias,
                            float* __restrict__ out) {
  const int lane = threadIdx.x & 31;
  const int wave = threadIdx.x >> 5;      // 0..7
  const int mt = wave & 1;                // row tile within block
  const int nt = wave >> 1;               // col tile 0..3
  const int m = lane & 15;
  const int kh = lane >> 4;               // 0: K=0,1   1: K=2,3
  const int rowBase = blockIdx.x * 32 + mt * 16;
  const int row = rowBase + m;
  const int k0 = 2 * kh;
  v2f a;
  a.x = (k0 < 3) ? xyz[(size_t)row * 3 + k0] : 0.0f;
  a.y = (k0 + 1 < 3) ? xyz[(size_t)row * 3 + k0 + 1] : 0.0f;
  const int col = nt * 16 + m;
  v2f bf;
  bf.x = (k0 < 3) ? W[k0 * 64 + col] : 0.0f;
  bf.y = (k0 + 1 < 3) ? W[(k0 + 1) * 64 + col] : 0.0f;
  v8f acc = {};
  acc = wmma_f32_4(a, bf, acc);
  const float bv = bias[col];
#pragma unroll
  for (int r = 0; r < 8; ++r) {
    const int orow = rowBase + r + kh * 8;
    out[(size_t)orow * 64 + col] = acc[r] + bv;
  }
}

// ---------------------------------------------------------------------------
// 2) Furthest point sampling. One block per batch. Entire point cloud AND the
//    running dmin array live in LDS (4*N floats; 256KB for N=16384 — fits the
//    CDNA5 320KB/WGP LDS). Writes the sampled centroid coordinates directly.
// ---------------------------------------------------------------------------
__global__ void fps_kernel(const float* __restrict__ pts,
                           float* __restrict__ newXyz, int N, int S) {
  extern __shared__ float smem[];
  float* sx = smem;
  float* sy = smem + N;
  float* sz = smem + 2 * N;
  float* dmin = smem + 3 * N;
  __shared__ float rbestv[32];
  __shared__ int rbesti[32];
  __shared__ int s_cur;

  const int b = blockIdx.x;
  const float* p = pts + (size_t)b * N * 3;
  for (int i = threadIdx.x; i < N; i += blockDim.x) {
    async_copy_b32(p + (size_t)i * 3 + 0, &sx[i]);
    async_copy_b32(p + (size_t)i * 3 + 1, &sy[i]);
    async_copy_b32(p + (size_t)i * 3 + 2, &sz[i]);
    dmin[i] = 1e10f;
  }
  if (threadIdx.x == 0) s_cur = 0;
  wait_async();
  __syncthreads();

  const int lane = threadIdx.x & 31;
  const int wav = threadIdx.x >> 5;
  const int nwaves = blockDim.x >> 5;

  for (int s = 0; s < S; ++s) {
    const int cur = s_cur;
    const float cx = sx[cur], cy = sy[cur], cz = sz[cur];
    if (threadIdx.x == 0) {
      float* o = newXyz + ((size_t)b * S + s) * 3;
      o[0] = cx; o[1] = cy; o[2] = cz;
    }
    float bestv = -1.0f;
    int besti = 0;
    for (int i = threadIdx.x; i < N; i += blockDim.x) {
      const float dx = sx[i] - cx, dy = sy[i] - cy, dz = sz[i] - cz;
      float dm = dmin[i];
      const float d = dx * dx + dy * dy + dz * dz;
      dm = fminf(dm, d);
      dmin[i] = dm;
      if (dm > bestv) { bestv = dm; besti = i; }
    }
    // wave32 argmax reduction
#pragma unroll
    for (int off = 16; off > 0; off >>= 1) {
      const float ov = __shfl_down(bestv, off, 32);
      const int oi = __shfl_down(besti, off, 32);
      if (ov > bestv) { bestv = ov; besti = oi; }
    }
    if (lane == 0) { rbestv[wav] = bestv; rbesti[wav] = besti; }
    __syncthreads();
    if (threadIdx.x == 0) {
      float bv = rbestv[0];
      int bi = rbesti[0];
      for (int w = 1; w < nwaves; ++w)
        if (rbestv[w] > bv) { bv = rbestv[w]; bi = rbesti[w]; }
      s_cur = bi;
    }
    __syncthreads();
  }
}

// ---------------------------------------------------------------------------
// 3) Sorted k-nearest (kmax=32) ball-query precursor. One block per batch;
//    source cloud cached in LDS (async staged); one thread per centroid does
//    an insertion select of the 32 smallest squared distances (ascending,
//    like the reference's top_k of -d2).
// ---------------------------------------------------------------------------
__global__ void knn_kernel(const float* __restrict__ src,
                           const float* __restrict__ qry,
                           int* __restrict__ nidx, float* __restrict__ ndist,
                           int N, int S) {
  extern __shared__ float smem[];
  float* sx = smem;
  float* sy = smem + N;
  float* sz = smem + 2 * N;
  const int b = blockIdx.x;
  const float* p = src + (size_t)b * N * 3;
  for (int i = threadIdx.x; i < N; i += blockDim.x) {
    async_copy_b32(p + (size_t)i * 3 + 0, &sx[i]);
    async_copy_b32(p + (size_t)i * 3 + 1, &sy[i]);
    async_copy_b32(p + (size_t)i * 3 + 2, &sz[i]);
  }
  wait_async();
  __syncthreads();
  for (int s = threadIdx.x; s < S; s += blockDim.x) {
    const float* q = qry + ((size_t)b * S + s) * 3;
    const float qx = q[0], qy = q[1], qz = q[2];
    float hd[KMAX];
    int hi[KMAX];
#pragma unroll
    for (int j = 0; j < KMAX; ++j) { hd[j] = 3.4e38f; hi[j] = 0; }
    float worst = 3.4e38f;
    for (int i = 0; i < N; ++i) {
      const float dx = sx[i] - qx, dy = sy[i] - qy, dz = sz[i] - qz;
      const float d = dx * dx + dy * dy + dz * dz;
      if (d < worst) {
        int j = KMAX - 1;
        while (j > 0 && hd[j - 1] > d) {
          hd[j] = hd[j - 1];
          hi[j] = hi[j - 1];
          --j;
        }
        hd[j] = d;
        hi[j] = i;
        worst = hd[KMAX - 1];
      }
    }
    int* oi = nidx + ((size_t)b * S + s) * KMAX;
    float* od = ndist + ((size_t)b * S + s) * KMAX;
#pragma unroll
    for (int j = 0; j < KMAX; ++j) { oi[j] = hi[j]; od[j] = hd[j]; }
  }
}

// ---------------------------------------------------------------------------
// 4) MSG MLP: gather [128 rows = RT centroids x k neighbors] x (cin+3) into
//    LDS (feature columns via async memory->LDS gather), GEMM1 (W1, relu) ->
//    LDS, GEMM2 (W2, relu) -> LDS, max-pool over k. All GEMM tiles are fp32
//    WMMA 16x16x4. Block = 512 threads (16 waves).
// ---------------------------------------------------------------------------
__global__ void mlp_kernel(const float* __restrict__ featIn,
                           const float* __restrict__ srcXyz,
                           const float* __restrict__ newXyz,
                           const int* __restrict__ nidx,
                           const float* __restrict__ ndist,
                           const float* __restrict__ W1,
                           const float* __restrict__ b1,
                           const float* __restrict__ W2,
                           const float* __restrict__ b2,
                           float* __restrict__ outFeat, int Nsrc, int S, int k,
                           int cin, int ch, float r2, int scaleOff,
                           int outStride) {
  extern __shared__ float smem[];
  const int cin3 = cin + 3;
  const int cpad = (cin3 + 3) & ~3;     // K padded to multiple of 4
  float* Hin = smem;                    // [128][cpad]
  float* H1 = smem + 128 * cpad;        // [128][ch]
  int* eidx = (int*)(H1 + 128 * ch);    // [128]
  const int b = blockIdx.y;
  const int RT = 128 / k;
  const int s0 = blockIdx.x * RT;
  const int tid = threadIdx.x;

  // Pull the (small, reused-by-every-block) weight matrices toward the WGP.
  __builtin_prefetch(W1, 0, 0);
  __builtin_prefetch(W2, 0, 0);

  // Effective neighbor index per row (ball-query mask: out-of-radius -> idx0)
  if (tid < 128) {
    const int t = tid / k, j = tid % k;
    const int s = s0 + t;
    const int* ip = nidx + ((size_t)b * S + s) * KMAX;
    const float* dp = ndist + ((size_t)b * S + s) * KMAX;
    int idx = ip[j];
    if (dp[j] > r2) idx = ip[0];
    eidx[tid] = idx;
  }
  __syncthreads();

  // Gather input tile: [rel_xyz(3) | feat(cin) | zero-pad].
  // Feature columns are an irregular gather -> use per-lane async
  // memory->LDS copies (each lane has its own global & LDS address).
  const int total = 128 * cpad;
  for (int e = tid; e < total; e += blockDim.x) {
    const int r = e / cpad, c = e - r * cpad;
    const int t = r / k;
    const int s = s0 + t;
    const int idx = eidx[r];
    if (c < 3) {
      Hin[r * cpad + c] = srcXyz[((size_t)b * Nsrc + idx) * 3 + c] -
                          newXyz[((size_t)b * S + s) * 3 + c];
    } else if (c < cin3) {
      async_copy_b32(&featIn[((size_t)b * Nsrc + idx) * cin + (c - 3)],
                     &Hin[r * cpad + c]);
    } else {
      Hin[r * cpad + c] = 0.0f;
    }
  }
  wait_async();
  __syncthreads();

  const int lane = tid & 31, wave = tid >> 5;  // 16 waves
  const int m = lane & 15, kh = lane >> 4;
  const int nColTiles = ch >> 4;
  const int nTiles = 8 * nColTiles;

  // GEMM1 + bias + relu -> H1
  for (int tile = wave; tile < nTiles; tile += 16) {
    const int mt = tile & 7, nt = tile >> 3;
    const int rowB = mt * 16, colB = nt * 16;
    v8f acc = {};
    for (int kk = 0; kk < cpad; kk += 4) {
      const int ka = kk + 2 * kh;
      v2f a, bb;
      a.x = Hin[(rowB + m) * cpad + ka];
      a.y = Hin[(rowB + m) * cpad + ka + 1];
      bb.x = (ka < cin3) ? W1[ka * ch + colB + m] : 0.0f;
      bb.y = (ka + 1 < cin3) ? W1[(ka + 1) * ch + colB + m] : 0.0f;
      acc = wmma_f32_4(a, bb, acc);
    }
    const float bv = b1[colB + m];
#pragma unroll
    for (int r = 0; r < 8; ++r) {
      const int orow = rowB + r + kh * 8;
      H1[orow * ch + colB + m] = fmaxf(acc[r] + bv, 0.0f);
    }
  }
  __syncthreads();

  // GEMM2 + bias + relu -> H2 (reuses Hin storage; cpad >= ch)
  float* H2 = Hin;
  for (int tile = wave; tile < nTiles; tile += 16) {
    const int mt = tile & 7, nt = tile >> 3;
    const int rowB = mt * 16, colB = nt * 16;
    v8f acc = {};
    for (int kk = 0; kk < ch; kk += 4) {
      const int ka = kk + 2 * kh;
      v2f a, bb;
      a.x = H1[(rowB + m) * ch + ka];
      a.y = H1[(rowB + m) * ch + ka + 1];
      bb.x = W2[ka * ch + colB + m];
      bb.y = W2[(ka + 1) * ch + colB + m];
      acc = wmma_f32_4(a, bb, acc);
    }
    const float bv = b2[colB + m];
#pragma unroll
    for (int r = 0; r < 8; ++r) {
      const int orow = rowB + r + kh * 8;
      H2[orow * ch + colB + m] = fmaxf(acc[r] + bv, 0.0f);
    }
  }
  __syncthreads();

  // Max-pool over k neighbors per centroid
  for (int e = tid; e < RT * ch; e += blockDim.x) {
    const int t = e / ch, c = e - t * ch;
    float mval = -3.4e38f;
    for (int j = 0; j < k; ++j) mval = fmaxf(mval, H2[(t * k + j) * ch + c]);
    outFeat[((size_t)b * S + (s0 + t)) * outStride + scaleOff + c] = mval;
  }
}

// ---------------------------------------------------------------------------
// 5) Global max+mean pool over S=512 centroids, then dense [512]x[512,256].
// ---------------------------------------------------------------------------
__global__ void pool_dense_kernel(const float* __restrict__ feat,
                                  const float* __restrict__ W,
                                  const float* __restrict__ bias,
                                  float* __restrict__ out, int S, int C) {
  __shared__ float gf[512];
  const int b = blockIdx.x;
  const int c = threadIdx.x;  // 256 threads, C == 256
  __builtin_prefetch(W, 0, 0);
  const float* f = feat + (size_t)b * S * C;
  float mv = -3.4e38f, sv = 0.0f;
  for (int s = 0; s < S; ++s) {
    const float v = f[(size_t)s * C + c];
    mv = fmaxf(mv, v);
    sv += v;
  }
  gf[c] = mv;
  gf[C + c] = sv / (float)S;
  __syncthreads();
  float acc = bias[c];
  for (int j = 0; j < 2 * C; ++j) acc += gf[j] * W[j * C + c];
  out[(size_t)b * C + c] = acc;
}

// ---------------------------------------------------------------------------
// Host orchestration
// ---------------------------------------------------------------------------
extern "C" void kernel_launch(void* const* d_in, const int* in_sizes, int n_in,
                              void* d_out, int out_size, void* d_ws,
                              size_t ws_size, hipStream_t stream) {
  (void)in_sizes; (void)n_in; (void)out_size; (void)ws_size;

  constexpr int B = 8, N = 16384, S0 = 2048, S1 = 512;

  const float* xyz   = (const float*)d_in[0];   // [B,N,3]
  const float* projW = (const float*)d_in[1];   // [3,64]
  const float* projB = (const float*)d_in[2];   // [64]
  // msg_params[0][0]: Ws[0],Ws[1],bs[0],bs[1]
  const float* L0S0W1 = (const float*)d_in[3];  // [67,64]
  const float* L0S0W2 = (const float*)d_in[4];  // [64,64]
  const float* L0S0B1 = (const float*)d_in[5];
  const float* L0S0B2 = (const float*)d_in[6];
  const float* L0S1W1 = (const float*)d_in[7];  // [67,64]
  const float* L0S1W2 = (const float*)d_in[8];
  const float* L0S1B1 = (const float*)d_in[9];
  const float* L0S1B2 = (const float*)d_in[10];
  const float* L1S0W1 = (const float*)d_in[11]; // [131,128]
  const float* L1S0W2 = (const float*)d_in[12]; // [128,128]
  const float* L1S0B1 = (const float*)d_in[13];
  const float* L1S0B2 = (const float*)d_in[14];
  const float* L1S1W1 = (const float*)d_in[15];
  const float* L1S1W2 = (const float*)d_in[16];
  const float* L1S1B1 = (const float*)d_in[17];
  const float* L1S1B2 = (const float*)d_in[18];
  const float* denseW = (const float*)d_in[19]; // [512,256]
  const float* denseB = (const float*)d_in[20]; // [256]

  // Workspace carve-up
  char* ws = (char*)d_ws;
  size_t off = 0;
  auto carve = [&](size_t bytes) {
    void* p = ws + off;
    off += (bytes + 255) & ~(size_t)255;
    return p;
  };
  float* feat0 = (float*)carve((size_t)B * N * 64 * 4);     // 33.6 MB
  float* nx0   = (float*)carve((size_t)B * S0 * 3 * 4);
  int*   nidx0 = (int*)carve((size_t)B * S0 * KMAX * 4);
  float* nd0   = (float*)carve((size_t)B * S0 * KMAX * 4);
  float* feat1 = (float*)carve((size_t)B * S0 * 128 * 4);   // 8.4 MB
  float* nx1   = (float*)carve((size_t)B * S1 * 3 * 4);
  int*   nidx1 = (int*)carve((size_t)B * S1 * KMAX * 4);
  float* nd1   = (float*)carve((size_t)B * S1 * KMAX * 4);
  float* feat2 = (float*)carve((size_t)B * S1 * 256 * 4);   // 4.2 MB

  // 1) projection: B*N rows, 32 rows per block
  proj_kernel<<<(B * N) / 32, 256, 0, stream>>>(xyz, projW, projB, feat0);

  // 2) FPS layer 0 (LDS: 4*N floats = 256KB per WG)
  fps_kernel<<<B, 1024, (size_t)4 * N * 4, stream>>>(xyz, nx0, N, S0);

  // 3) kNN layer 0 (LDS: 3*N floats = 192KB)
  knn_kernel<<<B, 1024, (size_t)3 * N * 4, stream>>>(xyz, nx0, nidx0, nd0, N,
                                                     S0);

  // 4) MSG MLPs layer 0 (cin=64 -> cpad=68, ch=64)
  {
    const size_t lds = (size_t)(128 * 68 + 128 * 64) * 4 + 128 * 4;
    dim3 g0(S0 / 8, B);   // k=16 -> RT=8
    mlp_kernel<<<g0, 512, lds, stream>>>(feat0, xyz, nx0, nidx0, nd0, L0S0W1,
                                         L0S0B1, L0S0W2, L0S0B2, feat1, N, S0,
                                         16, 64, 64, 0.01f, 0, 128);
    dim3 g1(S0 / 4, B);   // k=32 -> RT=4
    mlp_kernel<<<g1, 512, lds, stream>>>(feat0, xyz, nx0, nidx0, nd0, L0S1W1,
                                         L0S1B1, L0S1W2, L0S1B2, feat1, N, S0,
                                         32, 64, 64, 0.04f, 64, 128);
  }

  // 5) FPS layer 1 (sources = nx0, N=2048)
  fps_kernel<<<B, 1024, (size_t)4 * S0 * 4, stream>>>(nx0, nx1, S0, S1);

  // 6) kNN layer 1
  knn_kernel<<<B, 512, (size_t)3 * S0 * 4, stream>>>(nx0, nx1, nidx1, nd1, S0,
                                                     S1);

  // 7) MSG MLPs layer 1 (cin=128 -> cpad=132, ch=128; LDS = 133.6KB)
  {
    const size_t lds = (size_t)(128 * 132 + 128 * 128) * 4 + 128 * 4;
    dim3 g0(S1 / 8, B);   // k=16
    mlp_kernel<<<g0, 512, lds, stream>>>(feat1, nx0, nx1, nidx1, nd1, L1S0W1,
                                         L1S0B1, L1S0W2, L1S0B2, feat2, S0, S1,
                                         16, 128, 128, 0.04f, 0, 256);
    dim3 g1(S1 / 4, B);   // k=32
    mlp_kernel<<<g1, 512, lds, stream>>>(feat1, nx0, nx1, nidx1, nd1, L1S1W1,
                                         L1S1B1, L1S1W2, L1S1B2, feat2, S0, S1,
                                         32, 128, 128, 0.16f, 128, 256);
  }

  // 8) global pool + dense
  pool_dense_kernel<<<B, 256, 0, stream>>>(feat2, denseW, denseB,
                                           (float*)d_out, S1, 256);
}